// PottsModel_77146202571074
// MI455X (gfx1250) — compile-verified
//
#include <hip/hip_runtime.h>
#include <hip/hip_bf16.h>

// Potts clique energy: out[b] = sum_t coeffs[t] * (x[b, idx[t,0..3]] all equal)
// B=2048, N=4096, T=8192, K=4.
//
// MI455X strategy (v2):
//  - x values are exactly {0.0f,1.0f,2.0f}; their IEEE top bytes (0x00/0x3F/0x40)
//    are distinct, so float equality == equality of (bits>>24). Pack each row to
//    1 byte/elem in LDS: 16 rows = 64KB (stride 4100B -> conflict-free: bank =
//    m + idx/4 across the 16 M-lanes). Total LDS 107.6KB -> 3 workgroups/WGP
//    (vs 1 at f32 rows), 3x the latency hiding for the dependent idx->gather chain.
//  - T split 4-ways across blocks; per-block idx/coeff slices staged in LDS,
//    read as per-half-wave LDS broadcasts (b128 idx quads, b64 coeff pairs,
//    loaded unconditionally -> branchless v_cndmask, no saveexec in the loop).
//  - 'bt,t->b' contraction on the matrix core: V_WMMA_F32_16X16X4_F32, A[m,k] =
//    eq*coeff per the documented 16x4 f32 A layout, B = all-ones (layout-free),
//    f32 accumulation for 16 batches per wave.
//  - Cross-wave LDS reduce, then global_atomic_add_f32 across the 4 T-splits.

typedef float v2f __attribute__((ext_vector_type(2)));
typedef float v8f __attribute__((ext_vector_type(8)));

#define NN          4096                 // row length of x
#define ROWSTRIDE_B 4100                 // padded byte stride: (4100/4)%64==1 -> bank shift m
#define TILE_B      16                   // batch rows per block (== WMMA M)
#define T_BLOCK     2048                 // t-slice per block
#define WAVES       16                   // waves per block (512 threads, wave32)
#define T_WAVE      (T_BLOCK / WAVES)    // 128 t per wave -> 32 WMMAs

#define LDS_X_BYTES   (TILE_B * ROWSTRIDE_B)   // 65600 (16-aligned)
#define LDS_IDX_BYTES (T_BLOCK * 16)           // 32768
#define LDS_C_BYTES   (T_BLOCK * 4)            // 8192
#define LDS_RED_BYTES (WAVES * 16 * 4)         // 1024
#define LDS_TOTAL     (LDS_X_BYTES + LDS_IDX_BYTES + LDS_C_BYTES + LDS_RED_BYTES) // 107584

__device__ __forceinline__ float eq_gate(const unsigned char* __restrict__ xrow,
                                         int4 id, float c) {
  // 4 conflict-free ds_load_u8 gathers + branchless all-equal test.
  unsigned g0 = xrow[id.x];
  unsigned g1 = xrow[id.y];
  unsigned g2 = xrow[id.z];
  unsigned g3 = xrow[id.w];
  bool eq = (g0 == g1) & (g0 == g2) & (g0 == g3);
  return eq ? c : 0.0f;                  // c already in a VGPR -> v_cndmask, no exec games
}

__global__ void __launch_bounds__(512)
potts_wmma_kernel(const float* __restrict__ x,
                  const float* __restrict__ coeffs,
                  const int*   __restrict__ idx,
                  float*       __restrict__ out) {
  extern __shared__ char smem[];
  unsigned char* xsb  = (unsigned char*)smem;                   // TILE_B byte rows, padded
  int4*          idxs = (int4*)(smem + LDS_X_BYTES);            // T_BLOCK cliques
  float*         cs   = (float*)(smem + LDS_X_BYTES + LDS_IDX_BYTES);
  float*         red  = (float*)(smem + LDS_X_BYTES + LDS_IDX_BYTES + LDS_C_BYTES);

  const int tid    = threadIdx.x;
  const int b_base = blockIdx.x * TILE_B;
  const int t_base = blockIdx.y * T_BLOCK;

  // ---- Stage 16 contiguous x rows, packing f32 -> exponent byte (4x less LDS write BW) ----
  {
    const float4* xsrc = (const float4*)(x + (size_t)b_base * NN);
    #pragma unroll 4
    for (int i = tid; i < TILE_B * (NN / 4); i += 512) {   // 32 iters/thread, uniform
      int r = i >> 10;            // i / (NN/4)
      int c = i & 1023;           // i % (NN/4)
      float4 v = xsrc[i];
      unsigned p = (__float_as_uint(v.x) >> 24)
                 | ((__float_as_uint(v.y) >> 24) << 8)
                 | ((__float_as_uint(v.z) >> 24) << 16)
                 | (__float_as_uint(v.w) & 0xFF000000u);
      *(unsigned*)(xsb + r * ROWSTRIDE_B + c * 4) = p;     // 4B-aligned (4100%4==0)
    }
  }
  // ---- Stage idx slice (int4/clique) and coeff slice ----
  {
    const int4* isrc = (const int4*)(idx + (size_t)t_base * 4);
    #pragma unroll
    for (int i = tid; i < T_BLOCK; i += 512) idxs[i] = isrc[i];
    #pragma unroll
    for (int i = tid; i < T_BLOCK; i += 512) cs[i] = coeffs[t_base + i];
  }
  __syncthreads();

  // ---- Per-wave WMMA accumulation over its 128-t slice ----
  const int lane = tid & 31;
  const int wave = tid >> 5;
  const int m    = lane & 15;             // A-matrix row  (documented 16x4 f32 layout)
  const int kk   = (lane >> 4) << 1;      // lanes 0-15: K=0,1 ; lanes 16-31: K=2,3
  const unsigned char* xrow = xsb + m * ROWSTRIDE_B;

  v8f acc = {};
  v2f bones; bones.x = 1.0f; bones.y = 1.0f;   // B[k,n] = 1 everywhere: layout-independent

  const int tq0 = wave * T_WAVE;
  #pragma unroll 4
  for (int j = 0; j < T_WAVE; j += 4) {
    const int tA = tq0 + j + kk;          // even -> 8B-aligned coeff pair
    // Front-load this group's metadata: 2x ds_load_b128 + 1x ds_load_b64 (broadcast).
    int4   i0 = idxs[tA];
    int4   i1 = idxs[tA + 1];
    float2 cc = *(const float2*)(cs + tA);
    v2f a;
    a.x = eq_gate(xrow, i0, cc.x);        // A[m, kk]   -> VGPR0
    a.y = eq_gate(xrow, i1, cc.y);        // A[m, kk+1] -> VGPR1
    // D[m,n] += sum_k A[m,k] * 1 : full-f32 matrix-core accumulation
    acc = __builtin_amdgcn_wmma_f32_16x16x4_f32(
        /*neg_a=*/false, a, /*neg_b=*/false, bones,
        /*c_mod=*/(short)0, acc, /*reuse_a=*/false, /*reuse_b=*/false);
  }

  // ---- D layout: VGPR r, lanes 0-15 -> M=r ; lanes 16-31 -> M=r+8 (all N identical) ----
  if (lane == 0) {
    #pragma unroll
    for (int r = 0; r < 8; ++r) red[wave * 16 + r] = acc[r];
  } else if (lane == 16) {
    #pragma unroll
    for (int r = 0; r < 8; ++r) red[wave * 16 + 8 + r] = acc[r];
  }
  __syncthreads();

  // ---- Cross-wave reduce + atomic accumulate across the 4 T-splits ----
  if (tid < TILE_B) {
    float s = 0.0f;
    #pragma unroll
    for (int w = 0; w < WAVES; ++w) s += red[w * 16 + tid];
    atomicAdd(&out[b_base + tid], s);
  }
}

extern "C" void kernel_launch(void* const* d_in, const int* in_sizes, int n_in,
                              void* d_out, int out_size, void* d_ws, size_t ws_size,
                              hipStream_t stream) {
  const float* x      = (const float*)d_in[0];   // [B, N] f32
  const float* coeffs = (const float*)d_in[1];   // [T]    f32
  const int*   idx    = (const int*)d_in[2];     // [T, 4] i32
  float*       out    = (float*)d_out;           // [B]    f32

  const int B = out_size;            // 2048
  const int T = in_sizes[1];         // 8192

  // Allow >64KB dynamic LDS (host-side attribute; not a stream op, capture-safe).
  static bool attr_set = [] {
    (void)hipFuncSetAttribute((const void*)potts_wmma_kernel,
                              hipFuncAttributeMaxDynamicSharedMemorySize, LDS_TOTAL);
    return true;
  }();
  (void)attr_set;

  // T-split blocks accumulate via atomics -> out must start at zero every call.
  (void)hipMemsetAsync(d_out, 0, (size_t)out_size * sizeof(float), stream);

  dim3 grid(B / TILE_B, T / T_BLOCK);   // (128, 4)
  potts_wmma_kernel<<<grid, 512, LDS_TOTAL, stream>>>(x, coeffs, idx, out);
}